// StackedGCN_74672301408448
// MI455X (gfx1250) — compile-verified
//
#include <hip/hip_runtime.h>

#define N_NODES 100000
#define N_EDGES 1000000
#define DHID    64
#define KORD    8
#define NLAYERS 3

typedef float v2f __attribute__((ext_vector_type(2)));
typedef float v8f __attribute__((ext_vector_type(8)));

// ---------------------------------------------------------------------------
// Elementwise / graph kernels
// ---------------------------------------------------------------------------
__global__ void zero_kernel(float* __restrict__ p, int n) {
    int i = blockIdx.x * blockDim.x + threadIdx.x;
    int stride = gridDim.x * blockDim.x;
    for (; i < n; i += stride) p[i] = 0.0f;
}

__global__ void deg_kernel(const int* __restrict__ src, const int* __restrict__ dst,
                           const float* __restrict__ ew, float* __restrict__ deg, int nE) {
    int e = blockIdx.x * blockDim.x + threadIdx.x;
    if (e >= nE) return;
    int s = src[e];
    if (s == dst[e]) return;              // remove self loops
    atomicAdd(&deg[s], ew[e]);
}

__global__ void dinv_kernel(float* __restrict__ deg, int n) {
    int i = blockIdx.x * blockDim.x + threadIdx.x;
    if (i >= n) return;
    float d = deg[i];
    deg[i] = (d > 0.0f) ? rsqrtf(d) : 0.0f;
}

__global__ void norm_kernel(const int* __restrict__ src, const int* __restrict__ dst,
                            const float* __restrict__ ew, const float* __restrict__ dinv,
                            float* __restrict__ norm, int nE) {
    int e = blockIdx.x * blockDim.x + threadIdx.x;
    if (e >= nE) return;
    int s = src[e], d = dst[e];
    float w = (s == d) ? 0.0f : ew[e];
    norm[e] = -dinv[s] * w * dinv[d];
}

// out[dst] += norm_e * h[src]; 16 threads per edge, float4 per thread.
__global__ void scatter_kernel(const float* __restrict__ h, const int* __restrict__ src,
                               const int* __restrict__ dst, const float* __restrict__ norm,
                               float* __restrict__ out, int nE) {
    int t  = blockIdx.x * blockDim.x + threadIdx.x;
    int e  = t >> 4;
    int f4 = (t & 15) << 2;
    if (e >= nE) return;
    float w = norm[e];
    if (w == 0.0f) return;
    const float4 hv = *(const float4*)(h + (size_t)src[e] * DHID + f4);
    float* o = out + (size_t)dst[e] * DHID + f4;
    atomicAdd(o + 0, w * hv.x);
    atomicAdd(o + 1, w * hv.y);
    atomicAdd(o + 2, w * hv.z);
    atomicAdd(o + 3, w * hv.w);
}

// out = relu(acc + bias)
__global__ void relu_bias_kernel(const float* __restrict__ acc, const float* __restrict__ b,
                                 float* __restrict__ out, int n) {
    int i = blockIdx.x * blockDim.x + threadIdx.x;
    if (i >= n) return;
    out[i] = fmaxf(acc[i] + b[i & (DHID - 1)], 0.0f);
}

// ---------------------------------------------------------------------------
// WMMA GEMM:  acc[row,:64] (+)= T[row,:64] @ W[64,64]  (v_wmma_f32_16x16x4_f32)
//
// MODE 0: acc  = T @ W
// MODE 1: acc += T @ W
// MODE 2: T    = 2*T - t0 (Chebyshev recursion, written back);  acc += T @ W
//
// One wave -> 16 rows x 64 cols (4 v8f accumulators). 8 waves / block.
// A layout (16x4 f32): lanes 0-15 hold K=k0,k0+1 for M=lane; lanes 16-31 hold
// K=k0+2,k0+3 for M=lane-16. B layout (4x16) mirrors the K split with N=lane&15.
// W is staged in LDS K-pair-interleaved: lWp[(k/2)*128 + 2n + (k&1)] so each
// B fragment is a single aligned ds_load_b64 into an even VGPR pair.
// ---------------------------------------------------------------------------
template <int MODE>
__global__ void __launch_bounds__(256)
gemm64_wmma(float* __restrict__ T, const float* __restrict__ t0,
            const float* __restrict__ W, float* __restrict__ acc, int nrows) {
    __shared__ float lWp[DHID * DHID];
    for (int idx = threadIdx.x; idx < DHID * DHID; idx += 256) {
        int k = idx >> 6, n = idx & 63;
        lWp[((k >> 1) << 7) + (n << 1) + (k & 1)] = W[idx];
    }
    __syncthreads();

    const int wave = threadIdx.x >> 5;
    const int lane = threadIdx.x & 31;
    const int row0 = (blockIdx.x * 8 + wave) * 16;
    if (row0 >= nrows) return;               // wave-uniform: EXEC all-ones inside

    const int lrow  = lane & 15;
    const int khalf = (lane >> 4) << 1;      // 0 or 2
    const int crow  = row0 + ((lane >> 4) << 3);
    const int ccol  = lane & 15;

    v8f c0 = {}, c1 = {}, c2 = {}, c3 = {};
    if (MODE != 0) {
#pragma unroll
        for (int r = 0; r < 8; ++r) {
            const float* ap = acc + (size_t)(crow + r) * DHID + ccol;
            c0[r] = ap[0]; c1[r] = ap[16]; c2[r] = ap[32]; c3[r] = ap[48];
        }
    }

    float* arow       = T + (size_t)(row0 + lrow) * DHID + khalf;
    const float* trow = (MODE == 2) ? (t0 + (size_t)(row0 + lrow) * DHID + khalf) : nullptr;

#pragma unroll
    for (int kt = 0; kt < 16; ++kt) {
        const int k0 = kt * 4 + khalf;       // always even
        v2f a = *(const v2f*)(arow + kt * 4);
        if (MODE == 2) {
            v2f t = *(const v2f*)(trow + kt * 4);
            a.x = 2.0f * a.x - t.x;
            a.y = 2.0f * a.y - t.y;
            *(v2f*)(arow + kt * 4) = a;      // write Tk back for next recursion
        }
        const float* wp = lWp + ((k0 >> 1) << 7);
        v2f b;
        b = *(const v2f*)(wp + ((lrow +  0) << 1));
        c0 = __builtin_amdgcn_wmma_f32_16x16x4_f32(false, a, false, b, (short)0, c0, false, false);
        b = *(const v2f*)(wp + ((lrow + 16) << 1));
        c1 = __builtin_amdgcn_wmma_f32_16x16x4_f32(false, a, false, b, (short)0, c1, false, false);
        b = *(const v2f*)(wp + ((lrow + 32) << 1));
        c2 = __builtin_amdgcn_wmma_f32_16x16x4_f32(false, a, false, b, (short)0, c2, false, false);
        b = *(const v2f*)(wp + ((lrow + 48) << 1));
        c3 = __builtin_amdgcn_wmma_f32_16x16x4_f32(false, a, false, b, (short)0, c3, false, false);
    }

#pragma unroll
    for (int r = 0; r < 8; ++r) {
        float* ap = acc + (size_t)(crow + r) * DHID + ccol;
        ap[0] = c0[r]; ap[16] = c1[r]; ap[32] = c2[r]; ap[48] = c3[r];
    }
}

// ---------------------------------------------------------------------------
// Host orchestration
// ---------------------------------------------------------------------------
extern "C" void kernel_launch(void* const* d_in, const int* in_sizes, int n_in,
                              void* d_out, int out_size, void* d_ws, size_t ws_size,
                              hipStream_t stream) {
    const float* x       = (const float*)d_in[0];
    const int*   ei      = (const int*)d_in[1];
    const float* ew      = (const float*)d_in[2];
    const float* weights = (const float*)d_in[3];
    const float* bias    = (const float*)d_in[4];
    float*       out     = (float*)d_out;

    const int* src = ei;
    const int* dst = ei + N_EDGES;

    float* ws   = (float*)d_ws;
    float* norm = ws;                        // E
    float* dinv = ws + N_EDGES;              // N (degree, then D^-1/2 in place)
    float* base = ws + N_EDGES + N_NODES;
    float* P[5];
    for (int i = 0; i < 5; ++i) P[i] = base + (size_t)i * N_NODES * DHID;
    float* ACC = P[4];

    const int NF = N_NODES * DHID;
    dim3 blk(256);
    const int gE    = (N_EDGES + 255) / 256;
    const int gN    = (N_NODES + 255) / 256;
    const int gNF   = (NF + 255) / 256;
    const int gSc   = (N_EDGES * 16 + 255) / 256;
    const int gGemm = (N_NODES + 127) / 128;

    // --- scaled Laplacian edge norms ---
    zero_kernel<<<1024, blk, 0, stream>>>(dinv, N_NODES);
    deg_kernel <<<gE,   blk, 0, stream>>>(src, dst, ew, dinv, N_EDGES);
    dinv_kernel<<<gN,   blk, 0, stream>>>(dinv, N_NODES);
    norm_kernel<<<gE,   blk, 0, stream>>>(src, dst, ew, dinv, norm, N_EDGES);

    auto prop = [&](const float* tin, float* tout) {
        zero_kernel   <<<2048, blk, 0, stream>>>(tout, NF);
        scatter_kernel<<<gSc,  blk, 0, stream>>>(tin, src, dst, norm, tout, N_EDGES);
    };

    int h_idx = -1;                          // -1 => current h is input x
    for (int l = 0; l < NLAYERS; ++l) {
        // three rotation buffers distinct from the current h buffer
        int rid[3], c = 0;
        for (int i = 0; i < 4 && c < 3; ++i)
            if (i != h_idx) rid[c++] = i;

        const float* t0 = (h_idx < 0) ? x : P[h_idx];
        float* t1 = P[rid[0]];
        const float* Wl = weights + (size_t)l * KORD * DHID * DHID;

        // T0 = h, T1 = L_hat h ; acc = T0 W0 + T1 W1
        prop(t0, t1);
        gemm64_wmma<0><<<gGemm, blk, 0, stream>>>((float*)t0, nullptr, Wl,
                                                  ACC, N_NODES);
        gemm64_wmma<1><<<gGemm, blk, 0, stream>>>(t1, nullptr, Wl + DHID * DHID,
                                                  ACC, N_NODES);

        // free-buffer queue (indices into P); x (idx -1) is never pushed
        int q[10]; int qh = 0, qt = 0;
        q[qt++] = rid[1];
        q[qt++] = rid[2];
        int t0_idx = h_idx;
        int t1_idx = rid[0];

        for (int k = 2; k < KORD; ++k) {
            int s_idx = q[qh++];
            float* scat = P[s_idx];
            prop(P[t1_idx], scat);                         // scat = L_hat T(k-1)
            // fused: scat = 2*scat - T(k-2);  acc += scat @ Wk
            gemm64_wmma<2><<<gGemm, blk, 0, stream>>>(
                scat, (t0_idx < 0) ? x : P[t0_idx],
                Wl + (size_t)k * DHID * DHID, ACC, N_NODES);
            if (t0_idx >= 0) q[qt++] = t0_idx;
            t0_idx = t1_idx;
            t1_idx = s_idx;
        }

        // h_next = relu(acc + b); last-used rotation buffer is free to overwrite
        if (l == NLAYERS - 1) {
            relu_bias_kernel<<<gNF, blk, 0, stream>>>(ACC, bias + l * DHID, out, NF);
        } else {
            float* hn = P[t1_idx];            // content of last T no longer needed
            relu_bias_kernel<<<gNF, blk, 0, stream>>>(ACC, bias + l * DHID, hn, NF);
            h_idx = t1_idx;
        }
    }
}